// ParametricEqualizer_14276471292525
// MI455X (gfx1250) — compile-verified
//
#include <hip/hip_runtime.h>
#include <hip/hip_bf16.h>

// ParametricEqualizer on MI455X (gfx1250, wave32).
//
// out[m, f] = B(f)/A(f),  m in [0, 128*192), f in [0, 1025)
//   A(f) = a0 + a1 e^{-i th} + a2 e^{-i 2 th},  th = pi*f/1024   (same taps b -> B)
//
// Store-bandwidth bound: 201.5 MB out @ 23.3 TB/s ~ 8.7 us floor. The 3-tap
// zero-padded rfft is a K=3 GEMM done on the matrix pipe with
// V_WMMA_F32_16X16X4_F32. Per 16-row x 16-freq tile: 4 WMMAs
// (a-taps/b-taps x cos-part/-sin-part twiddles), so every lane owns one
// frequency column with Dre/Dim/Nre/Nim all lane-local -> no cross-lane
// shuffles, complex divide with one v_rcp_f32, contiguous float2 (b64) stores.

typedef __attribute__((ext_vector_type(2))) float v2f;
typedef __attribute__((ext_vector_type(8))) float v8f;

#define NROWS (128 * 192)        // 24576 rows (batch*filters)
#define NFREQ 1025               // 2048/2 + 1
#define F2STRIDE NFREQ           // row stride in float2 (complex) units
#define MTILES (NROWS / 16)      // 1536
#define FTILES 65                // ceil(1025/16); tile 64 = Nyquist tail (n==0 only)
#define FT_PER_WAVE 4
#define CHUNKS ((FTILES + FT_PER_WAVE - 1) / FT_PER_WAVE)  // 17

// ---------------------------------------------------------------- taps kernel
__global__ __launch_bounds__(256)
void eq_taps_kernel(const float* __restrict__ center,
                    const float* __restrict__ gain,
                    const float* __restrict__ q,
                    float* __restrict__ taps) {
  int i = blockIdx.x * 256 + threadIdx.x;
  if (i >= NROWS) return;
  const float kTwoPiOverSR = 6.28318530717958647692f / 44100.0f;
  float w0 = kTwoPiOverSR * center[i];
  float A  = __expf(gain[i] * 0.05756462732485114210f);  // ln(10)/40
  float s, c;
  __sincosf(w0, &s, &c);
  float alpha = s * 0.5f * __builtin_amdgcn_rcpf(q[i]);
  float aA  = alpha * A;                          // numerator (b) taps
  float adA = alpha * __builtin_amdgcn_rcpf(A);   // denominator (a) taps
  float m2c = -2.0f * c;
  float* t = taps + i * 6;
  t[0] = 1.0f + adA;  // a0
  t[1] = m2c;         // a1
  t[2] = 1.0f - adA;  // a2
  t[3] = 1.0f + aA;   // b0
  t[4] = m2c;         // b1
  t[5] = 1.0f - aA;   // b2
}

// ------------------------------------------------------- combine + store tile
__device__ __forceinline__
void eq_combine_store(const v8f& Dre, const v8f& Dim,
                      const v8f& Nre, const v8f& Nim,
                      float2* __restrict__ base, bool pred) {
#pragma unroll
  for (int v = 0; v < 8; ++v) {
    float dr = Dre[v], di = Dim[v], nr = Nre[v], ni = Nim[v];
    float inv = __builtin_amdgcn_rcpf(__builtin_fmaf(dr, dr, di * di));
    float re  = __builtin_fmaf(nr, dr, ni * di) * inv;
    float im  = __builtin_fmaf(ni, dr, -nr * di) * inv;
    if (pred) base[v * F2STRIDE] = make_float2(re, im);
  }
}

// ---------------------------------------------------------------- main kernel
// One wave per (16-row M-tile, chunk of FT_PER_WAVE 16-freq tiles).
__global__ __launch_bounds__(256)
void eq_main_kernel(const float* __restrict__ taps, float* __restrict__ out) {
  const int lane = threadIdx.x & 31;
  const int wib  = threadIdx.x >> 5;             // wave in block (8 waves)
  const int task = blockIdx.x * 8 + wib;         // wave-uniform
  const int mt    = task / CHUNKS;
  const int chunk = task % CHUNKS;
  if (mt >= MTILES) return;                      // wave-uniform guard

  const int m16  = lane & 15;
  const int half = lane >> 4;                    // 0: lanes 0-15, 1: lanes 16-31
  const int n    = lane & 15;                    // D column = local freq index

  // Load this wave's 16 rows of taps (halves mirror rows).
  const float* tp = taps + (mt * 16 + m16) * 6;
  float a0 = tp[0], a1 = tp[1], a2 = tp[2];
  float b0 = tp[3], b1 = tp[4], b2 = tp[5];

  // A-matrix 16x4 f32 layout (ISA 7.12.2):
  //   VGPR0: lanes 0-15 -> K=0, lanes 16-31 -> K=2
  //   VGPR1: lanes 0-15 -> K=1, lanes 16-31 -> K=3 (zero pad)
  v2f Aden, Anum;
  Aden.x = half ? a2 : a0;
  Aden.y = half ? 0.0f : a1;
  Anum.x = half ? b2 : b0;
  Anum.y = half ? 0.0f : b1;

  for (int t = 0; t < FT_PER_WAVE; ++t) {
    int ft = chunk * FT_PER_WAVE + t;            // wave-uniform
    if (ft >= FTILES) break;
    int f0   = ft * 16;
    int freq = f0 + n;

    // Twiddle B-matrices 4x16 (column n = freq f0+n), split into real and
    // imaginary parts of e^{-i K th}:
    //   Bcos row K = cos(K th) :  K0=1, K1=cos th, K2=cos 2th, K3=0
    //   Bsin row K = -sin(K th):  K0=0, K1=-sin th, K2=-sin 2th, K3=0
    // B layout mirrors A: VGPR0 = rows K0(lanes 0-15)/K2(16-31), VGPR1 = K1/K3.
    float theta = 0.00306796157577128245f * (float)freq;  // pi/1024 * f
    float s, co;
    __sincosf(theta, &s, &co);
    float c2 = __builtin_fmaf(2.0f * co, co, -1.0f);      // cos 2th
    float s2 = 2.0f * s * co;                             // sin 2th
    v2f Bcos, Bsin;
    Bcos.x = half ? c2   : 1.0f;
    Bcos.y = half ? 0.0f : co;
    Bsin.x = half ? -s2  : 0.0f;
    Bsin.y = half ? 0.0f : -s;

    v8f Dre = {0.f,0.f,0.f,0.f,0.f,0.f,0.f,0.f};
    v8f Dim = {0.f,0.f,0.f,0.f,0.f,0.f,0.f,0.f};
    v8f Nre = {0.f,0.f,0.f,0.f,0.f,0.f,0.f,0.f};
    v8f Nim = {0.f,0.f,0.f,0.f,0.f,0.f,0.f,0.f};
    Dre = __builtin_amdgcn_wmma_f32_16x16x4_f32(false, Aden, false, Bcos,
                                                (short)0, Dre, false, false);
    Dim = __builtin_amdgcn_wmma_f32_16x16x4_f32(false, Aden, false, Bsin,
                                                (short)0, Dim, false, false);
    Nre = __builtin_amdgcn_wmma_f32_16x16x4_f32(false, Anum, false, Bcos,
                                                (short)0, Nre, false, false);
    Nim = __builtin_amdgcn_wmma_f32_16x16x4_f32(false, Anum, false, Bsin,
                                                (short)0, Nim, false, false);

    // D layout: VGPR v -> row (mt*16 + half*8 + v), column n -> freq f0+n.
    // Lane stores an (re,im) float2 -> lanes 0-15 cover 128 contiguous bytes.
    float2* base = (float2*)out + (size_t)(mt * 16 + half * 8) * F2STRIDE + freq;
    if (ft != FTILES - 1) {
      eq_combine_store(Dre, Dim, Nre, Nim, base, true);       // full tile
    } else {
      eq_combine_store(Dre, Dim, Nre, Nim, base, n == 0);     // Nyquist tail
    }
  }
}

// ---------------------------------------------------------------- launch
extern "C" void kernel_launch(void* const* d_in, const int* in_sizes, int n_in,
                              void* d_out, int out_size, void* d_ws, size_t ws_size,
                              hipStream_t stream) {
  (void)in_sizes; (void)n_in; (void)out_size; (void)ws_size;
  const float* center = (const float*)d_in[0];
  const float* gain   = (const float*)d_in[1];
  const float* q      = (const float*)d_in[2];
  float* out  = (float*)d_out;
  float* taps = (float*)d_ws;   // NROWS * 6 floats = 576 KB scratch

  eq_taps_kernel<<<(NROWS + 255) / 256, 256, 0, stream>>>(center, gain, q, taps);

  int tasks = MTILES * CHUNKS;                   // 26112 wave tasks
  eq_main_kernel<<<(tasks + 7) / 8, 256, 0, stream>>>(taps, out);
}